// EEGGATNet_64888365908227
// MI455X (gfx1250) — compile-verified
//
#include <hip/hip_runtime.h>
#include <math.h>
#include <stdint.h>

typedef __attribute__((ext_vector_type(2))) float v2f;
typedef __attribute__((ext_vector_type(8))) float v8f;

#define NEG_SLOPE 0.2f

__device__ __forceinline__ float leakyf(float x) { return x > 0.f ? x : NEG_SLOPE * x; }
__device__ __forceinline__ float eluf(float x)   { return x > 0.f ? x : __expf(x) - 1.f; }

// float atomic max via int/uint atomics (works with -inf init)
__device__ __forceinline__ void atomicMaxF(float* addr, float v) {
    if (v >= 0.f) atomicMax((int*)addr, __float_as_int(v));
    else          atomicMin((unsigned int*)addr, __float_as_uint(v));
}

// ---------------------------------------------------------------------------
// C[M,Nn] = A[M,K] @ B[K,Nn]  fp32 WMMA 16x16x4, LDS-staged B via async loads.
//
// Block = 128 threads = 4 waves. Block tile: 64(M) x 64(N).
//   wave w -> rows [blockIdx.y*64 + w*16, +16), all 64 cols (4 accumulators).
// B k-slab (KB=32 x 64) staged in LDS with GLOBAL_LOAD_ASYNC_TO_LDS_B128
// (ASYNCcnt), fenced by s_wait_asynccnt 0 + workgroup barrier, then read back
// with ds_load. LDS row stride 80 floats => half-wave rows (K, K+2) are 32
// banks apart: conflict-free.
// A fragment (v2f per lane) loaded from global and reused for 4 WMMAs.
// Requires K % 32 == 0 and Nn % 64 == 0 (true here: K=128/256, Nn=256/64).
// ---------------------------------------------------------------------------
#define GKB 32
#define GBN 64
#define GSTR 80   // LDS row stride in floats (80*4 = 320B, 16B aligned)

__global__ void gemm_f32_wmma(const float* __restrict__ A,
                              const float* __restrict__ B,
                              float* __restrict__ C,
                              int M, int K, int Nn) {
    __shared__ __attribute__((aligned(16))) float sB[GKB * GSTR];

    const int t    = threadIdx.x;
    const int lane = t & 31;
    const int wave = t >> 5;
    const int half = lane >> 4;     // 0: K pair {k,k+1}, 1: {k+2,k+3}
    const int l15  = lane & 15;

    const int cb = blockIdx.x * GBN;          // col base of block tile
    const int mb = blockIdx.y * 64;           // row base of block tile

    int r = mb + wave * 16 + l15;             // this lane's A row
    if (r >= M) r = M - 1;                    // clamp; stores are guarded
    const float* Arow = A + (size_t)r * K;

    const unsigned sB_base = (unsigned)(size_t)&sB[0];

    // cooperative async-load mapping: 512 float4 = 32 rows x 16 float4/row
    const int c4  = t & 15;                   // float4 column within slab
    const int rw0 = t >> 4;                   // base row (0..7), step 8

    v8f acc0 = {}, acc1 = {}, acc2 = {}, acc3 = {};

    for (int kk = 0; kk < K; kk += GKB) {
#pragma unroll
        for (int j = 0; j < 4; ++j) {
            const int row = rw0 + j * 8;      // 0..31
            const unsigned ldsoff = sB_base + (unsigned)(row * GSTR + c4 * 4) * 4u;
            const uint64_t gaddr  = (uint64_t)(B + (size_t)(kk + row) * Nn + cb + c4 * 4);
            asm volatile("global_load_async_to_lds_b128 %0, %1, off"
                         :: "v"(ldsoff), "v"(gaddr) : "memory");
        }
        asm volatile("s_wait_asynccnt 0" ::: "memory");
        __syncthreads();

        for (int ks = 0; ks < GKB; ks += 4) {
            const int ka = kk + ks + half * 2;      // global k for A fragment
            v2f a = *(const v2f*)(Arow + ka);
            const int kl = ks + half * 2;           // local row in sB
#pragma unroll
            for (int nt = 0; nt < 4; ++nt) {
                v2f b;
                b.x = sB[(kl + 0) * GSTR + nt * 16 + l15];
                b.y = sB[(kl + 1) * GSTR + nt * 16 + l15];
                v8f* pacc = (nt == 0) ? &acc0 : (nt == 1) ? &acc1 : (nt == 2) ? &acc2 : &acc3;
                *pacc = __builtin_amdgcn_wmma_f32_16x16x4_f32(
                    false, a, false, b, (short)0, *pacc, false, false);
            }
        }
        __syncthreads();
    }

    const int rb = mb + wave * 16 + half * 8;
#pragma unroll
    for (int nt = 0; nt < 4; ++nt) {
        const v8f acc = (nt == 0) ? acc0 : (nt == 1) ? acc1 : (nt == 2) ? acc2 : acc3;
        const int c = cb + nt * 16 + l15;
#pragma unroll
        for (int j = 0; j < 8; ++j) {
            const int row = rb + j;
            if (row < M) C[(size_t)row * Nn + c] = acc[j];
        }
    }
}

// es[n,h] = <h[n,h,:], a_src[h,:]> ; ed likewise
__global__ void node_alpha(const float* __restrict__ h,
                           const float* __restrict__ a_src,
                           const float* __restrict__ a_dst,
                           float* __restrict__ es, float* __restrict__ ed,
                           int N, int H, int D) {
    int t = blockIdx.x * blockDim.x + threadIdx.x;
    if (t >= N * H) return;
    const int n = t / H, hh = t % H;
    const float* hp = h + (size_t)n * H * D + (size_t)hh * D;
    const float* as = a_src + hh * D;
    const float* ad = a_dst + hh * D;
    float s0 = 0.f, s1 = 0.f;
    for (int i = 0; i < D; ++i) { float v = hp[i]; s0 += v * as[i]; s1 += v * ad[i]; }
    es[t] = s0; ed[t] = s1;
}

__global__ void fill_f32(float* __restrict__ p, float v, long long n) {
    long long t = (long long)blockIdx.x * blockDim.x + threadIdx.x;
    if (t < n) p[t] = v;
}

// pass 1: segment max of edge logits over dst (self-loops: t in [E, E+N))
__global__ void edge_max(const int* __restrict__ src, const int* __restrict__ dst,
                         int E, int N,
                         const float* __restrict__ es, const float* __restrict__ ed,
                         float* __restrict__ m, int H) {
    int t = blockIdx.x * blockDim.x + threadIdx.x;
    const int Et = E + N;
    if (t >= Et) return;
    const int s = (t < E) ? src[t] : (t - E);
    const int d = (t < E) ? dst[t] : (t - E);
    for (int h = 0; h < H; ++h) {
        float e = leakyf(es[s * H + h] + ed[d * H + h]);
        atomicMaxF(&m[d * H + h], e);
    }
}

// pass 2: denominator sum of exp(e - m[dst])
__global__ void edge_sum(const int* __restrict__ src, const int* __restrict__ dst,
                         int E, int N,
                         const float* __restrict__ es, const float* __restrict__ ed,
                         const float* __restrict__ m, float* __restrict__ ssum, int H) {
    int t = blockIdx.x * blockDim.x + threadIdx.x;
    const int Et = E + N;
    if (t >= Et) return;
    const int s = (t < E) ? src[t] : (t - E);
    const int d = (t < E) ? dst[t] : (t - E);
    for (int h = 0; h < H; ++h) {
        float e = leakyf(es[s * H + h] + ed[d * H + h]);
        atomicAdd(&ssum[d * H + h], __expf(e - m[d * H + h]));
    }
}

// pass 3: out[dst,h,:] += alpha * h[src,h,:]  (thread per (edge, head), float4 gather)
__global__ void edge_aggr(const int* __restrict__ src, const int* __restrict__ dst,
                          int E, int N,
                          const float* __restrict__ es, const float* __restrict__ ed,
                          const float* __restrict__ m, const float* __restrict__ ssum,
                          const float* __restrict__ hfeat, float* __restrict__ out,
                          int H, int D) {
    long long t = (long long)blockIdx.x * blockDim.x + threadIdx.x;
    const long long total = (long long)(E + N) * H;
    if (t >= total) return;
    const int eidx = (int)(t / H);
    const int h    = (int)(t % H);
    const int s = (eidx < E) ? src[eidx] : (eidx - E);
    const int d = (eidx < E) ? dst[eidx] : (eidx - E);
    float e   = leakyf(es[s * H + h] + ed[d * H + h]);
    float den = ssum[d * H + h];
    den = den > 0.f ? den : 1.f;
    const float alpha = __expf(e - m[d * H + h]) / den;
    const float4* hp4 = (const float4*)(hfeat + (size_t)s * H * D + (size_t)h * D);
    float*        op  = out + (size_t)d * H * D + (size_t)h * D;
    const int D4 = D >> 2;
    for (int f = 0; f < D4; ++f) {
        float4 v = hp4[f];
        atomicAdd(&op[4 * f + 0], alpha * v.x);
        atomicAdd(&op[4 * f + 1], alpha * v.y);
        atomicAdd(&op[4 * f + 2], alpha * v.z);
        atomicAdd(&op[4 * f + 3], alpha * v.w);
    }
}

__global__ void bias_elu(float* __restrict__ x, const float* __restrict__ b,
                         int N, int C) {
    long long t = (long long)blockIdx.x * blockDim.x + threadIdx.x;
    if (t >= (long long)N * C) return;
    const int c = (int)(t % C);
    x[t] = eluf(x[t] + b[c]);
}

__global__ void pool_sum(const float* __restrict__ h, const int* __restrict__ batch,
                         float* __restrict__ sums, float* __restrict__ cnt,
                         int N, int C) {
    long long t = (long long)blockIdx.x * blockDim.x + threadIdx.x;
    if (t >= (long long)N * C) return;
    const int n = (int)(t / C);
    const int c = (int)(t % C);
    const int g = batch[n];
    atomicAdd(&sums[(size_t)g * C + c], h[t]);
    if (c == 0) atomicAdd(&cnt[g], 1.0f);
}

__global__ void classify(const float* __restrict__ sums, const float* __restrict__ cnt,
                         const float* __restrict__ Wc, const float* __restrict__ bc,
                         float* __restrict__ out, int G, int Cin, int Cout) {
    int t = blockIdx.x * blockDim.x + threadIdx.x;
    if (t >= G * Cout) return;
    const int g = t / Cout, c = t % Cout;
    float cc = cnt[g]; cc = cc > 1.f ? cc : 1.f;
    float acc = bc[c];
    const float inv = 1.f / cc;
    for (int k = 0; k < Cin; ++k)
        acc += (sums[(size_t)g * Cin + k] * inv) * Wc[k * Cout + c];
    out[t] = acc;
}

static inline int cdiv(long long a, long long b) { return (int)((a + b - 1) / b); }

extern "C" void kernel_launch(void* const* d_in, const int* in_sizes, int n_in,
                              void* d_out, int out_size, void* d_ws, size_t ws_size,
                              hipStream_t stream) {
    const float* x      = (const float*)d_in[0];
    const int*   eidx   = (const int*)  d_in[1];
    const int*   batch  = (const int*)  d_in[2];
    // d_in[3] = num_graphs scalar on device: unused (G derived from out_size)
    const float* W1     = (const float*)d_in[4];
    const float* a_src1 = (const float*)d_in[5];
    const float* a_dst1 = (const float*)d_in[6];
    const float* b1     = (const float*)d_in[7];
    const float* W2     = (const float*)d_in[8];
    const float* a_src2 = (const float*)d_in[9];
    const float* a_dst2 = (const float*)d_in[10];
    const float* b2     = (const float*)d_in[11];
    const float* Wc     = (const float*)d_in[12];
    const float* bc     = (const float*)d_in[13];
    float* out = (float*)d_out;

    const int HD1 = in_sizes[5];            // H*D  (= 256)
    const int F1  = in_sizes[4] / HD1;      // F_IN (= 128)
    const int N   = in_sizes[0] / F1;       // nodes (= 50000)
    const int E   = in_sizes[1] / 2;        // edges (= 1e6)
    const int HID = in_sizes[9];            // D per head (= 64), layer2 width
    const int H1  = HD1 / HID;              // heads (= 4)
    const int NC  = in_sizes[13];           // classes (= 2)
    const int G   = out_size / NC;          // graphs (= 512)
    const int Et  = E + N;                  // edges incl. self-loops

    const int* src = eidx;
    const int* dst = eidx + E;

    // ---- workspace layout (floats) ----
    float* ws = (float*)d_ws;
    size_t o = 0;
    float* h1   = ws + o; o += (size_t)N * HD1;   // x@W1
    float* out1 = ws + o; o += (size_t)N * HD1;   // layer1 aggregation / activation
    float* es1  = ws + o; o += (size_t)N * H1;
    float* ed1  = ws + o; o += (size_t)N * H1;
    float* m1   = ws + o; o += (size_t)N * H1;
    float* s1   = ws + o; o += (size_t)N * H1;
    float* h2   = ws + o; o += (size_t)N * HID;   // out1@W2
    float* out2 = ws + o; o += (size_t)N * HID;
    float* es2  = ws + o; o += (size_t)N;
    float* ed2  = ws + o; o += (size_t)N;
    float* m2   = ws + o; o += (size_t)N;
    float* s2   = ws + o; o += (size_t)N;
    float* sums = ws + o; o += (size_t)G * HID;
    float* cntg = ws + o; o += (size_t)G;
    (void)ws_size; (void)n_in;

    const int TB = 256;
    const int ebl = cdiv(Et, TB);

    // ================= Layer 1 =================
    {   // h1 = x @ W1   [N,F1]x[F1,HD1]
        dim3 grid(cdiv(HD1, GBN), cdiv(N, 64));
        gemm_f32_wmma<<<grid, 128, 0, stream>>>(x, W1, h1, N, F1, HD1);
    }
    node_alpha<<<cdiv((long long)N * H1, TB), TB, 0, stream>>>(h1, a_src1, a_dst1, es1, ed1, N, H1, HID);
    fill_f32<<<cdiv((long long)N * H1, TB), TB, 0, stream>>>(m1, -INFINITY, (long long)N * H1);
    fill_f32<<<cdiv((long long)N * H1, TB), TB, 0, stream>>>(s1, 0.f, (long long)N * H1);
    fill_f32<<<cdiv((long long)N * HD1, TB), TB, 0, stream>>>(out1, 0.f, (long long)N * HD1);
    edge_max <<<ebl, TB, 0, stream>>>(src, dst, E, N, es1, ed1, m1, H1);
    edge_sum <<<ebl, TB, 0, stream>>>(src, dst, E, N, es1, ed1, m1, s1, H1);
    edge_aggr<<<cdiv((long long)Et * H1, TB), TB, 0, stream>>>(src, dst, E, N, es1, ed1, m1, s1, h1, out1, H1, HID);
    bias_elu <<<cdiv((long long)N * HD1, TB), TB, 0, stream>>>(out1, b1, N, HD1);

    // ================= Layer 2 =================
    {   // h2 = out1 @ W2   [N,HD1]x[HD1,HID]
        dim3 grid(cdiv(HID, GBN), cdiv(N, 64));
        gemm_f32_wmma<<<grid, 128, 0, stream>>>(out1, W2, h2, N, HD1, HID);
    }
    node_alpha<<<cdiv((long long)N, TB), TB, 0, stream>>>(h2, a_src2, a_dst2, es2, ed2, N, 1, HID);
    fill_f32<<<cdiv((long long)N, TB), TB, 0, stream>>>(m2, -INFINITY, (long long)N);
    fill_f32<<<cdiv((long long)N, TB), TB, 0, stream>>>(s2, 0.f, (long long)N);
    fill_f32<<<cdiv((long long)N * HID, TB), TB, 0, stream>>>(out2, 0.f, (long long)N * HID);
    edge_max <<<ebl, TB, 0, stream>>>(src, dst, E, N, es2, ed2, m2, 1);
    edge_sum <<<ebl, TB, 0, stream>>>(src, dst, E, N, es2, ed2, m2, s2, 1);
    edge_aggr<<<cdiv((long long)Et, TB), TB, 0, stream>>>(src, dst, E, N, es2, ed2, m2, s2, h2, out2, 1, HID);
    bias_elu <<<cdiv((long long)N * HID, TB), TB, 0, stream>>>(out2, b2, N, HID);

    // ================= Pool + classifier =================
    fill_f32<<<cdiv((long long)G * HID, TB), TB, 0, stream>>>(sums, 0.f, (long long)G * HID);
    fill_f32<<<cdiv((long long)G, TB), TB, 0, stream>>>(cntg, 0.f, (long long)G);
    pool_sum<<<cdiv((long long)N * HID, TB), TB, 0, stream>>>(out2, batch, sums, cntg, N, HID);
    classify<<<cdiv((long long)G * NC, TB), TB, 0, stream>>>(sums, cntg, Wc, bc, out, G, HID, NC);
}